// ScanConv_77429670412661
// MI455X (gfx1250) — compile-verified
//
#include <hip/hip_runtime.h>
#include <math.h>

// Problem constants (from reference): B,C,N,S,OUT = 16,64,2048,20,64
#define BB   16
#define CC   64
#define NN   2048
#define SS   20
#define OUTC 64
#define KTOT (CC * SS)   // 1280

typedef __attribute__((ext_vector_type(2))) float v2f;
typedef __attribute__((ext_vector_type(8))) float v8f;
typedef unsigned int u32x4 __attribute__((ext_vector_type(4)));
typedef int          i32x4 __attribute__((ext_vector_type(4)));
typedef int          i32x8 __attribute__((ext_vector_type(8)));

// ---------------------------------------------------------------------------
// TDM helper: async 2-D tile load (global -> LDS), f32 elements.
// Descriptor per CDNA5 ISA ch.8 (Tensor DMA Descriptor D#).
// Tracked by TENSORcnt; caller does s_wait_tensorcnt + barrier.
// ---------------------------------------------------------------------------
__device__ __attribute__((always_inline)) void tdm_load_2d_f32(
        unsigned lds_byte_off, const float* gaddr,
        unsigned tensor_d0, unsigned tensor_d1,
        unsigned tile_d0, unsigned tile_d1,
        unsigned long long stride0_elems) {
    unsigned long long ga = (unsigned long long)(uintptr_t)gaddr;
    u32x4 g0;
    g0.x = 1u;                                              // count=1, user desc
    g0.y = lds_byte_off;                                    // lds_addr
    g0.z = (unsigned)(ga & 0xFFFFFFFFu);                    // global_addr[31:0]
    g0.w = (unsigned)((ga >> 32) & 0x1FFFFFFu) | (2u << 30);// addr[56:32] | type=2
    i32x8 g1;
    g1[0] = (int)(2u << 16);                                // data_size=4B
    g1[1] = (int)((tensor_d0 & 0xFFFFu) << 16);             // tensor_dim0 lo
    g1[2] = (int)((tensor_d0 >> 16) | ((tensor_d1 & 0xFFFFu) << 16));
    g1[3] = (int)((tensor_d1 >> 16) | (tile_d0 << 16));     // | tile_dim0
    g1[4] = (int)(tile_d1 & 0xFFFFu);                       // tile_dim1 (dim2=0)
    g1[5] = (int)(stride0_elems & 0xFFFFFFFFull);           // dim0 stride lo
    g1[6] = (int)((stride0_elems >> 32) & 0xFFFFull);       // dim0 stride hi
    g1[7] = 0;
    i32x4 z4 = {0, 0, 0, 0};
#if __clang_major__ >= 23
    i32x8 z8 = {0, 0, 0, 0, 0, 0, 0, 0};
    __builtin_amdgcn_tensor_load_to_lds(g0, g1, z4, z4, z8, 0);
#else
    __builtin_amdgcn_tensor_load_to_lds(g0, g1, z4, z4, 0);
#endif
}

__device__ __attribute__((always_inline)) unsigned lds_offset_of(const void* p) {
    // generic pointer to LDS: low 32 bits are the LDS byte offset
    return (unsigned)(unsigned long long)(uintptr_t)p;
}

// ---------------------------------------------------------------------------
// 1) Squared norms: sqn[b*N+n] = sum_c x[b,c,n]^2
// ---------------------------------------------------------------------------
__global__ void sqnorm_kernel(const float* __restrict__ x, float* __restrict__ sqn) {
    int t = blockIdx.x * blockDim.x + threadIdx.x;   // t = b*N + n
    if (t >= BB * NN) return;
    int b = t >> 11;
    int n = t & (NN - 1);
    const float* xb = x + (size_t)b * CC * NN + n;
    float s = 0.f;
    #pragma unroll
    for (int c = 0; c < CC; ++c) {
        float v = xb[c * NN];
        s += v * v;
    }
    sqn[t] = s;
}

// ---------------------------------------------------------------------------
// 2) W pre-transpose: Wt[k*OUT + o] = W[o*KTOT + k]  (K-major for WMMA B frags)
// ---------------------------------------------------------------------------
__global__ void wt_kernel(const float* __restrict__ W, float* __restrict__ Wt) {
    int t = blockIdx.x * blockDim.x + threadIdx.x;   // t = k*OUT + o
    if (t >= KTOT * OUTC) return;
    int k = t / OUTC;
    int o = t - k * OUTC;
    Wt[t] = W[(size_t)o * KTOT + k];
}

// ---------------------------------------------------------------------------
// 3) KNN via fp32 WMMA Gram + double-buffered TDM tile staging.
//    Block = 256 thr (8 waves) owns 128 rows of one batch.
//    xA (64c x 128n) staged once by TDM, reused across all m-chunks.
//    Per 32-col chunk: TDM for chunk i+1 is issued right after chunk i is
//    published, overlapping DMA with the 32 WMMAs + selection of chunk i.
//    Selection: 128 threads fold Gram cols (two passes of 16) into
//    register-resident top-20 lists (stable -> matches jax top_k order).
// ---------------------------------------------------------------------------
__global__ void __launch_bounds__(256) knn_kernel(const float* __restrict__ x,
                                                  const float* __restrict__ sqn,
                                                  int* __restrict__ idxg) {
    __shared__ float xA[CC * 128];       // 32 KB  [c][row]
    __shared__ float xB[2][CC * 32];     // 16 KB  double-buffered [c][col]
    __shared__ float dtile[128 * 16];    //  8 KB  [row][col16] Gram pass tile
    __shared__ float sm_s[2][32];

    const int tid  = threadIdx.x;
    const int w    = tid >> 5;
    const int lane = tid & 31;
    const int hi   = (lane >= 16) ? 1 : 0;
    const int b    = blockIdx.y;
    const int n0   = blockIdx.x * 128;
    const float* xbase = x + (size_t)b * CC * NN;
    const int nchunks = NN / 32;         // 64

    // prologue: stage xA (once) + first xB chunk
    if (tid < 32) {
        tdm_load_2d_f32(lds_offset_of(xA), xbase + n0, NN, CC, 128, CC, NN);
        tdm_load_2d_f32(lds_offset_of(xB[0]), xbase, NN, CC, 32, CC, NN);
        sm_s[0][tid] = sqn[b * NN + tid];
    }

    // per-thread top-20 (threads 0..127 own one row each)
    float sn = 0.f;
    if (tid < 128) sn = sqn[b * NN + n0 + tid];
    float td[SS];
    int   ti[SS];
    #pragma unroll
    for (int k = 0; k < SS; ++k) { td[k] = -INFINITY; ti[k] = 0; }

    const int arow = w * 16 + (lane & 15);
    const int ncol = lane & 15;

    for (int i = 0; i < nchunks; ++i) {
        const int buf = i & 1;
        const int m0  = i * 32;
        if (tid < 32) __builtin_amdgcn_s_wait_tensorcnt(0);
        __syncthreads();                                  // chunk i published

        // overlap: kick DMA for chunk i+1 into the other buffer
        if (tid < 32 && (i + 1) < nchunks) {
            tdm_load_2d_f32(lds_offset_of(xB[buf ^ 1]), xbase + (m0 + 32),
                            NN, CC, 32, CC, NN);
            sm_s[buf ^ 1][tid] = sqn[b * NN + m0 + 32 + tid];
        }

        // ---- Gram sub-tile: wave w -> rows [w*16, w*16+16) x 32 cols ----
        const float* xb = xB[buf];
        v8f acc0 = {};
        v8f acc1 = {};
        #pragma unroll
        for (int kt = 0; kt < 16; ++kt) {
            int k = kt * 4 + hi * 2;      // fp32 16x16x4 A/B K layout
            v2f a;
            a.x = xA[k * 128 + arow];
            a.y = xA[(k + 1) * 128 + arow];
            v2f b0, b1;
            b0.x = xb[k * 32 + ncol];
            b0.y = xb[(k + 1) * 32 + ncol];
            b1.x = xb[k * 32 + ncol + 16];
            b1.y = xb[(k + 1) * 32 + ncol + 16];
            acc0 = __builtin_amdgcn_wmma_f32_16x16x4_f32(
                false, a, false, b0, (short)0, acc0, false, false);
            acc1 = __builtin_amdgcn_wmma_f32_16x16x4_f32(
                false, a, false, b1, (short)0, acc1, false, false);
        }

        // ---- two selection passes (cols 0-15, then 16-31) ----
        #pragma unroll
        for (int pass = 0; pass < 2; ++pass) {
            #pragma unroll
            for (int r = 0; r < 8; ++r) {
                int row = w * 16 + r + hi * 8;
                dtile[row * 16 + ncol] = pass ? acc1[r] : acc0[r];
            }
            __syncthreads();
            if (tid < 128) {
                const float* dr = &dtile[tid * 16];
                const int mbase = m0 + pass * 16;
                for (int j = 0; j < 16; ++j) {
                    float d = 2.f * dr[j] - sn - sm_s[buf][pass * 16 + j];
                    if (d > td[SS - 1]) {
                        td[SS - 1] = d;
                        ti[SS - 1] = mbase + j;
                        // static-index bubble-up stays in VGPRs; strict '>'
                        // keeps ties stable (lower m first, like jax top_k)
                        #pragma unroll
                        for (int q = SS - 1; q > 0; --q) {
                            if (td[q] > td[q - 1]) {
                                float tv = td[q]; td[q] = td[q - 1]; td[q - 1] = tv;
                                int   tt = ti[q]; ti[q] = ti[q - 1]; ti[q - 1] = tt;
                            }
                        }
                    }
                }
            }
            __syncthreads();
        }
    }

    if (tid < 128) {
        int* op = idxg + (size_t)(b * NN + n0 + tid) * SS;
        #pragma unroll
        for (int k = 0; k < SS; ++k) op[k] = ti[k];
    }
}

// ---------------------------------------------------------------------------
// 4) Fused gather + adjweight (20x20) + ELU + GEMM(e @ W^T) + bias + transpose
//    Block: 128 threads = 4 waves, 64 points. Wave owns 16 points (WMMA M).
//    K = C*S = 1280 walked in 16 channel-chunks of 4 (80 k, 20 WMMA steps).
//    W chunk (80k x 64out, 20 KB) staged by TDM, issued BEFORE the gather/ELU
//    phase so the DMA hides behind that VALU-heavy work.
// ---------------------------------------------------------------------------
__global__ void __launch_bounds__(128) fused_gemm_kernel(
        const float* __restrict__ x, const float* __restrict__ adjw,
        const int* __restrict__ idxg, const float* __restrict__ Wt,
        const float* __restrict__ bias, float* __restrict__ out) {
    __shared__ float adj_s[SS * SS];        // 1.6 KB
    __shared__ int   idx_s[64 * SS];        // 5 KB
    __shared__ float e_s[4 * 16 * 80];      // 20 KB: [wave][p(16)][k_loc(80)]
    __shared__ float wt_s[80 * OUTC];       // 20 KB: [k_loc][o]

    const int tid  = threadIdx.x;
    const int w    = tid >> 5;
    const int lane = tid & 31;
    const int hi   = (lane >= 16) ? 1 : 0;
    const int g0   = blockIdx.x * 64;       // first global point (b*N+n)
    const int b    = g0 >> 11;
    const int n0   = g0 & (NN - 1);

    for (int i = tid; i < SS * SS; i += 128) adj_s[i] = adjw[i];
    for (int i = tid; i < 64 * SS; i += 128) idx_s[i] = idxg[(size_t)g0 * SS + i];
    __syncthreads();

    v8f acc[4] = {};   // 4 output tiles of 16 cols -> OUT=64

    for (int c0 = 0; c0 < CC; c0 += 4) {
        // ---- kick W-chunk DMA (contiguous 80x64 f32 = 20 KB) ----
        if (tid < 32) {
            tdm_load_2d_f32(lds_offset_of(wt_s),
                            Wt + (size_t)(c0 * SS) * OUTC,
                            80 * OUTC, 1, 80 * OUTC, 1, 80 * OUTC);
        }

        // ---- phase 1 (overlaps DMA): e_s[w][p][c_loc*20+s] ----
        #pragma unroll
        for (int it = 0; it < 2; ++it) {
            int item = it * 32 + lane;      // (p, c_loc) in 16x4
            int p  = item >> 2;
            int cl = item & 3;
            int c  = c0 + cl;
            const float* xc  = x + ((size_t)b * CC + c) * NN;
            const int*   ids = &idx_s[(w * 16 + p) * SS];
            float nb[SS];
            #pragma unroll
            for (int k = 0; k < SS; ++k) nb[k] = xc[ids[k]];
            float* ep = &e_s[(w * 16 + p) * 80 + cl * SS];
            #pragma unroll 4
            for (int s = 0; s < SS; ++s) {
                float t = 0.f;
                #pragma unroll
                for (int k = 0; k < SS; ++k) t += nb[k] * adj_s[k * SS + s];
                ep[s] = (t > 0.f) ? t : expm1f(t);   // elu, alpha=1
            }
        }
        if (tid < 32) __builtin_amdgcn_s_wait_tensorcnt(0);
        __syncthreads();

        // ---- phase 2: 20 steps of v_wmma_f32_16x16x4_f32, all from LDS ----
        const float* erow = &e_s[(w * 16 + (lane & 15)) * 80];
        #pragma unroll 2
        for (int kt = 0; kt < 20; ++kt) {
            int kl = kt * 4 + hi * 2;       // fp32 A/B 16x16x4 K layout
            v2f a;
            a.x = erow[kl];
            a.y = erow[kl + 1];
            const float* wp = &wt_s[kl * OUTC + (lane & 15)];
            #pragma unroll
            for (int ot = 0; ot < 4; ++ot) {
                v2f bf;
                bf.x = wp[ot * 16];           // B[k,   ncol]
                bf.y = wp[OUTC + ot * 16];    // B[k+1, ncol]
                acc[ot] = __builtin_amdgcn_wmma_f32_16x16x4_f32(
                    false, a, false, bf, (short)0, acc[ot], false, false);
            }
        }
        __syncthreads();
    }

    // ---- epilogue: bias + store transposed out[b, o, n] ----
    const int ncol = lane & 15;
    float bv[4];
    #pragma unroll
    for (int ot = 0; ot < 4; ++ot) bv[ot] = bias[ot * 16 + ncol];
    #pragma unroll
    for (int r = 0; r < 8; ++r) {
        int row = r + hi * 8;               // C/D layout: VGPR r -> M = r (+8 hi)
        int n   = n0 + w * 16 + row;
        #pragma unroll
        for (int ot = 0; ot < 4; ++ot) {
            int o = ot * 16 + ncol;
            out[((size_t)b * OUTC + o) * NN + n] = acc[ot][r] + bv[ot];
        }
    }
}

// ---------------------------------------------------------------------------
// Launch
// ---------------------------------------------------------------------------
extern "C" void kernel_launch(void* const* d_in, const int* in_sizes, int n_in,
                              void* d_out, int out_size, void* d_ws, size_t ws_size,
                              hipStream_t stream) {
    const float* x    = (const float*)d_in[0];   // (B,C,N)
    const float* adjw = (const float*)d_in[1];   // (S,S)
    const float* W    = (const float*)d_in[2];   // (OUT, C*S)
    const float* bias = (const float*)d_in[3];   // (OUT,)
    // d_in[4] = spiral_size scalar; S=20 baked in at compile time.
    float* out = (float*)d_out;                  // (B,OUT,N)

    // Workspace layout (~3 MB total)
    float* sqn  = (float*)d_ws;                                            // B*N f32
    int*   idxg = (int*)((char*)d_ws + (size_t)BB * NN * 4);               // B*N*S i32
    float* Wt   = (float*)((char*)d_ws + (size_t)BB * NN * 4
                                       + (size_t)BB * NN * SS * 4);        // KTOT*OUT f32

    sqnorm_kernel<<<(BB * NN + 255) / 256, 256, 0, stream>>>(x, sqn);
    wt_kernel<<<(KTOT * OUTC + 255) / 256, 256, 0, stream>>>(W, Wt);

    dim3 kgrid(NN / 128, BB);
    knn_kernel<<<kgrid, 256, 0, stream>>>(x, sqn, idxg);

    fused_gemm_kernel<<<(BB * NN) / 64, 128, 0, stream>>>(x, adjw, idxg, Wt, bias, out);
}